// CapsuleLayer_35828617183257
// MI455X (gfx1250) — compile-verified
//
#include <hip/hip_runtime.h>
#include <math.h>

typedef float v2f __attribute__((ext_vector_type(2)));
typedef float v8f __attribute__((ext_vector_type(8)));

constexpr int B_   = 256;
constexpr int IU   = 8;
constexpr int IC   = 1152;
constexpr int NU   = 10;
constexpr int US   = 16;
constexpr int K2   = IC * IU;   // 9216 contraction length for s_j
constexpr int SROW = NU * US;   // 160

// ---------------- data prep ----------------
// Xt[b][k] = x[b,i,c] (row-major [B_ x K2], k contiguous)  -- A-side of s_j GEMM
// Xc[k][b] = x[b,i,c] (row-major [K2 x B_], b contiguous)  -- A-side of agree GEMM
__global__ void prep_x(const float* __restrict__ x,
                       float* __restrict__ xt, float* __restrict__ xc) {
  int idx = blockIdx.x * blockDim.x + threadIdx.x;   // exact: B_*K2 threads
  int b = idx / K2, k = idx - b * K2;
  int c = k >> 3, i = k & 7;
  float val = x[(b * IU + i) * IC + c];
  xt[idx] = val;
  xc[(size_t)k * B_ + b] = val;
}

// Wt[d][u][k] = W0[c,d,u,i]  with k = c*8 + i   (k contiguous)
__global__ void prep_wt(const float* __restrict__ w, float* __restrict__ wt) {
  int idx = blockIdx.x * blockDim.x + threadIdx.x;   // exact: NU*US*K2 threads
  int d = idx / (US * K2), rem = idx - d * (US * K2);
  int u = rem / K2, k = rem - u * K2;
  int c = k >> 3, i = k & 7;
  wt[idx] = w[((c * NU + d) * US + u) * IU + i];
}

__global__ void zero_b(float* __restrict__ b) {
  int i = blockIdx.x * blockDim.x + threadIdx.x;
  if (i < NU * IC) b[i] = 0.f;
}

// ---------------- softmax over channels (axis 0 of b_ij[c,d]) ----------------
// b_ij / c_ij stored as [d][c] for contiguous per-d reductions.
__global__ void softmax_c(const float* __restrict__ bij, float* __restrict__ cij) {
  int d = blockIdx.x;
  int t = threadIdx.x;
  __shared__ float red[256];
  const float* bd = bij + d * IC;
  float m = -1e30f;
  for (int c = t; c < IC; c += 256) m = fmaxf(m, bd[c]);
  red[t] = m; __syncthreads();
  for (int s = 128; s > 0; s >>= 1) { if (t < s) red[t] = fmaxf(red[t], red[t + s]); __syncthreads(); }
  float mx = red[0]; __syncthreads();
  float sum = 0.f;
  for (int c = t; c < IC; c += 256) sum += __expf(bd[c] - mx);
  red[t] = sum; __syncthreads();
  for (int s = 128; s > 0; s >>= 1) { if (t < s) red[t] += red[t + s]; __syncthreads(); }
  float inv = 1.f / red[0];
  for (int c = t; c < IC; c += 256) cij[d * IC + c] = __expf(bd[c] - mx) * inv;
}

// Ws[d][u][k] = Wt[d][u][k] * c_ij[d][k>>3]   (re-done each routing iteration)
__global__ void scale_w(const float* __restrict__ wt, const float* __restrict__ cij,
                        float* __restrict__ wsc) {
  int idx = blockIdx.x * blockDim.x + threadIdx.x;   // exact: NU*US*K2 threads
  int d = idx / (US * K2), rem = idx - d * (US * K2);
  int k = rem % K2;
  wsc[idx] = wt[idx] * cij[d * IC + (k >> 3)];
}

// ---------------- s_j GEMM (per d): [16b x K2] @ [K2 x 16u] ------------------
// 8 waves split K=9216; inner loop = two float2 loads + one WMMA.
// Cross-wave accumulate through LDS. u_hat never materialized.
__global__ void __launch_bounds__(256) sj_gemm(const float* __restrict__ xt,
                                               const float* __restrict__ wsc,
                                               float* __restrict__ s) {
  int btile = blockIdx.x;            // 0..15
  int d     = blockIdx.y;            // 0..9
  int lane  = threadIdx.x & 31;
  int wave  = threadIdx.x >> 5;      // 0..7
  int half  = lane >> 4;             // K sub-pair select per ISA A/B layout
  int ln    = lane & 15;

  const float* arow = xt  + (size_t)(btile * 16 + ln) * K2;   // A row (m = ln)
  const float* brow = wsc + (size_t)(d * US + ln) * K2;       // B col (n = ln), k-contig

  v8f acc = {};
  int kbeg = wave * (K2 / 8), kend = kbeg + (K2 / 8);
  for (int k0 = kbeg; k0 < kend; k0 += 4) {
    int kk = k0 + half * 2;
    v2f a = *(const v2f*)(arow + kk);   // A[m][kk], A[m][kk+1]
    v2f b = *(const v2f*)(brow + kk);   // B[kk][n], B[kk+1][n]
    acc = __builtin_amdgcn_wmma_f32_16x16x4_f32(false, a, false, b,
                                                (short)0, acc, false, false);
  }
  __shared__ float red[8][256];
#pragma unroll
  for (int r = 0; r < 8; ++r) red[wave][lane * 8 + r] = acc[r];
  __syncthreads();
  int t = threadIdx.x;
  float sum = 0.f;
#pragma unroll
  for (int w = 0; w < 8; ++w) sum += red[w][t];
  int l2 = t >> 3, r2 = t & 7;
  int m = r2 + ((l2 >> 4) << 3);                 // C/D layout: VGPR r, lane half
  int n = l2 & 15;
  s[(btile * 16 + m) * SROW + d * US + n] = sum;
}

// ---------------- squash (reference squashes over num_units axis!) -----------
// Emits v in [b][d][u] (output layout) and vt in [d][u][b] (agree B-side).
__global__ void squash_k(const float* __restrict__ s, float* __restrict__ v,
                         float* __restrict__ vt) {
  int t = blockIdx.x * blockDim.x + threadIdx.x;   // exact: B_*US
  int b = t >> 4, u = t & 15;
  const float* sb = s + b * SROW + u;
  float msq = 0.f;
#pragma unroll
  for (int d = 0; d < NU; ++d) { float e = sb[d * US]; msq += e * e; }
  float scale = sqrtf(msq) / (1.f + msq);          // msq/(1+msq) * 1/sqrt(msq)
  float* vb = v + b * SROW + u;
#pragma unroll
  for (int d = 0; d < NU; ++d) {
    float val = sb[d * US] * scale;
    vb[d * US] = val;
    vt[(size_t)(d * US + u) * B_ + b] = val;
  }
}

// ---------------- agree GEMM + fused epilogue --------------------------------
// Per (d, ktile): D[16k x 16u] = Xc-tile @ vt-tile accumulated over b (K=256),
// then agree[c,d] += (1/B) * sum_{k in tile, u} Wt[d][u][k] * D[k][u],
// reduced in-wave (wave32 shfl_xor) and atomically added into b_ij.
__global__ void __launch_bounds__(256) agree_gemm(const float* __restrict__ xc,
                                                  const float* __restrict__ vt,
                                                  const float* __restrict__ wt,
                                                  float* __restrict__ bij) {
  int d     = blockIdx.y;
  int wave  = threadIdx.x >> 5;
  int ktile = blockIdx.x * 8 + wave;   // 0..575
  int kbase = ktile * 16;
  int lane  = threadIdx.x & 31;
  int half  = lane >> 4, ln = lane & 15;

  const float* arow = xc + (size_t)(kbase + ln) * B_;       // A row (m = k = ln)
  const float* brow = vt + (size_t)(d * US + ln) * B_;      // B col (n = u = ln)

  v8f acc = {};
  for (int b0 = 0; b0 < B_; b0 += 4) {
    int j = b0 + half * 2;
    v2f a = *(const v2f*)(arow + j);    // A'[m][j], A'[m][j+1]
    v2f b = *(const v2f*)(brow + j);    // B'[j][n], B'[j+1][n]
    acc = __builtin_amdgcn_wmma_f32_16x16x4_f32(false, a, false, b,
                                                (short)0, acc, false, false);
  }
  // acc[r] = M[d][kbase + r + half*8][u = ln]
  const float* wrow = wt + (size_t)(d * US + ln) * K2 + kbase + half * 8;
  float p = 0.f;
#pragma unroll
  for (int r = 0; r < 8; ++r) p += acc[r] * wrow[r];   // 8 consecutive floats
  // reduce across the 16 lanes of each half (rows 0-7 vs 8-15 -> c0 vs c0+1)
  p += __shfl_xor(p, 1, 32);
  p += __shfl_xor(p, 2, 32);
  p += __shfl_xor(p, 4, 32);
  p += __shfl_xor(p, 8, 32);
  if (ln == 0) {
    int c = (kbase >> 3) + half;
    atomicAdd(&bij[d * IC + c], p * (1.f / 256.f));
  }
}

// ---------------- host ----------------
extern "C" void kernel_launch(void* const* d_in, const int* in_sizes, int n_in,
                              void* d_out, int out_size, void* d_ws, size_t ws_size,
                              hipStream_t stream) {
  const float* x = (const float*)d_in[0];   // [256, 8, 1152]
  const float* W = (const float*)d_in[1];   // [1, 1152, 10, 16, 8]
  float* ws  = (float*)d_ws;
  float* xt  = ws;                                   // B_*K2      = 2,359,296 f
  float* xc  = xt  + (size_t)B_ * K2;                // K2*B_      = 2,359,296 f
  float* wt  = xc  + (size_t)K2 * B_;                // NU*US*K2   = 1,474,560 f
  float* wsc = wt  + (size_t)NU * US * K2;           // NU*US*K2   = 1,474,560 f
  float* s   = wsc + (size_t)NU * US * K2;           // B_*SROW    =    40,960 f
  float* v   = s   + (size_t)B_ * SROW;              // B_*SROW    =    40,960 f
  float* vt  = v   + (size_t)B_ * SROW;              // NU*US*B_   =    40,960 f
  float* bij = vt  + (size_t)NU * US * B_;           // NU*IC      =    11,520 f
  float* cij = bij + (size_t)NU * IC;                // NU*IC      =    11,520 f
  // total ~31.3 MB of f32 workspace

  prep_x<<<(B_ * K2) / 256, 256, 0, stream>>>(x, xt, xc);
  prep_wt<<<(NU * US * K2) / 256, 256, 0, stream>>>(W, wt);
  zero_b<<<(NU * IC + 255) / 256, 256, 0, stream>>>(bij);

  for (int it = 0; it < 3; ++it) {
    softmax_c<<<NU, 256, 0, stream>>>(bij, cij);
    scale_w<<<(NU * US * K2) / 256, 256, 0, stream>>>(wt, cij, wsc);
    sj_gemm<<<dim3(B_ / 16, NU), 256, 0, stream>>>(xt, wsc, s);
    float* vout = (it == 2) ? (float*)d_out : v;
    squash_k<<<(B_ * US) / 256, 256, 0, stream>>>(s, vout, vt);
    if (it < 2)
      agree_gemm<<<dim3(K2 / 16 / 8, NU), 256, 0, stream>>>(xc, vt, wt, bij);
  }
}